// YourClass_27350351741024
// MI455X (gfx1250) — compile-verified
//
#include <hip/hip_runtime.h>
#include <math.h>

// ---------------- problem constants ----------------
#define H_SZ   768
#define P_SZ   256
#define I_DIM  80
#define B_SZ   256
#define T_LEN  400
#define G4H    (4 * H_SZ)           // 3072 gate columns

// ---------------- launch configuration ----------------
#define NUM_WG 96                   // 4 m-groups x 24 n-groups (gate GEMM)
#define TPB    256                  // 8 wave32s per workgroup

typedef __attribute__((ext_vector_type(2))) float v2f;
typedef __attribute__((ext_vector_type(8))) float v8f;

__device__ __forceinline__ float sigmoidf_(float x) {
    return 1.0f / (1.0f + __expf(-x));
}

__device__ __forceinline__ v8f wmma_f32(v2f a, v2f b, v8f c) {
    return __builtin_amdgcn_wmma_f32_16x16x4_f32(false, a, false, b,
                                                 (short)0, c, false, false);
}

// Device-wide barrier for a persistent grid (all NUM_WG blocks co-resident).
__device__ __forceinline__ void grid_barrier(unsigned* count, unsigned* gen) {
    __threadfence();                 // release this thread's global writes
    __syncthreads();
    if (threadIdx.x == 0) {
        unsigned my   = __hip_atomic_load(gen, __ATOMIC_RELAXED, __HIP_MEMORY_SCOPE_AGENT);
        unsigned prev = __hip_atomic_fetch_add(count, 1u, __ATOMIC_ACQ_REL, __HIP_MEMORY_SCOPE_AGENT);
        if (prev == NUM_WG - 1) {
            __hip_atomic_store(count, 0u, __ATOMIC_RELAXED, __HIP_MEMORY_SCOPE_AGENT);
            __hip_atomic_fetch_add(gen, 1u, __ATOMIC_RELEASE, __HIP_MEMORY_SCOPE_AGENT);
        } else {
            while (__hip_atomic_load(gen, __ATOMIC_ACQUIRE, __HIP_MEMORY_SCOPE_AGENT) == my) {
                __builtin_amdgcn_s_sleep(2);
            }
        }
    }
    __syncthreads();
    __threadfence();                 // acquire: see other WGs' writes
}

// Zero recurrent state + barrier words (must run every launch for determinism).
__global__ void init_kernel(float* c, float* h, unsigned* bars) {
    int idx = blockIdx.x * blockDim.x + threadIdx.x;
    if (idx < B_SZ * H_SZ) c[idx] = 0.0f;
    if (idx < B_SZ * P_SZ) h[idx] = 0.0f;
    if (idx < 4)           bars[idx] = 0u;
}

// One full LSTMP layer: persistent grid, sequential over T with grid barriers.
// Phase 1: gates = [x_t | h] @ [Wx ; Wh]   (A staged in LDS, fp32 WMMA)
// Phase 2: c = sig(f)*c + sig(i)*tanh(g) ; g2 = sig(o)*tanh(c)
// Phase 3: h = g2 @ Wp  -> h_state and seq_out[:, t, :]
template <int IN_DIM>
__global__ __launch_bounds__(TPB) void lstmp_layer_kernel(
    const float* __restrict__ seq_in,   // [B, T, IN_DIM]
    const float* __restrict__ Wx,       // [IN_DIM, 4H]
    const float* __restrict__ Wh,       // [P, 4H]
    const float* __restrict__ bias,     // [4H]
    const float* __restrict__ Wp,       // [H, P]
    float* __restrict__ seq_out,        // [B, T, P]
    float* __restrict__ c_state,        // [B, H]
    float* __restrict__ h_state,        // [B, P]
    float* __restrict__ gates,          // [B, 4H] scratch
    float* __restrict__ g2,             // [B, H] scratch
    unsigned* bar_count, unsigned* bar_gen)
{
    extern __shared__ float ldsA[];     // [64][KPAD] staged activations
    constexpr int KTOT = IN_DIM + P_SZ; // 336 or 512
    constexpr int KPAD = KTOT + 4;      // bank-conflict padding, 16B aligned
    constexpr int KV   = KTOT / 4;      // float4 chunks per row

    const int lane = threadIdx.x & 31;
    const int wave = threadIdx.x >> 5;
    const int l15  = lane & 15;         // tile row (A) / tile col (B, C)
    const int kgrp = lane >> 4;         // selects K pair {0,1} vs {2,3}
    const int mhi  = (lane >= 16) ? 8 : 0;

    // Gate GEMM: WG block = 64 rows x 128 cols; wave sub-block = 32x32 (2m x 2n tiles)
    const int mg     = blockIdx.x / 24;      // 0..3
    const int ng     = blockIdx.x % 24;      // 0..23
    const int m_base = mg * 64;
    const int n_base = ng * 128;
    const int wm     = wave >> 2;            // 0..1
    const int wn     = wave & 3;             // 0..3
    const int mrow0  = wm * 32;              // row offset inside WG block
    const int ncol0  = n_base + wn * 32;     // global col base for this wave

    // Projection tile: 16x16 tiles over [256 x 256] -> first 256 waves
    const int gwave = blockIdx.x * 8 + wave;
    const int pm = (gwave < 256) ? (gwave >> 4) * 16 : -1;
    const int pn = (gwave & 15) * 16;

    for (int t = 0; t < T_LEN; ++t) {
        // ---------- Stage A = [x_t | h] rows m_base..m_base+63 into LDS ----------
        for (int idx = threadIdx.x; idx < 64 * KV; idx += TPB) {
            const int r  = idx / KV;
            const int c  = (idx - r * KV) * 4;
            const int gm = m_base + r;
            float4 v;
            if (c < IN_DIM) {
                v = *(const float4*)(seq_in + (size_t)gm * T_LEN * IN_DIM
                                            + (size_t)t * IN_DIM + c);
            } else {
                v = *(const float4*)(h_state + (size_t)gm * P_SZ + (c - IN_DIM));
            }
            *(float4*)(&ldsA[(size_t)r * KPAD + c]) = v;
        }
        __syncthreads();

        // ---------- Phase 1: gate GEMM (2m x 2n tiles per wave) ----------
        {
            v8f a00 = {}, a01 = {}, a10 = {}, a11 = {};
            const float* lrow0 = ldsA + (size_t)(mrow0 + l15) * KPAD;
            const float* lrow1 = lrow0 + (size_t)16 * KPAD;
            const int nb = ncol0 + l15;
#pragma unroll 2
            for (int k0 = 0; k0 < IN_DIM; k0 += 4) {
                const int kk = k0 + 2 * kgrp;
                const v2f av0 = *(const v2f*)(lrow0 + kk);
                const v2f av1 = *(const v2f*)(lrow1 + kk);
                const float* br = Wx + (size_t)kk * G4H;
                v2f b0; b0.x = br[nb];      b0.y = br[nb + G4H];
                v2f b1; b1.x = br[nb + 16]; b1.y = br[nb + 16 + G4H];
                a00 = wmma_f32(av0, b0, a00);
                a01 = wmma_f32(av0, b1, a01);
                a10 = wmma_f32(av1, b0, a10);
                a11 = wmma_f32(av1, b1, a11);
            }
#pragma unroll 2
            for (int k0 = 0; k0 < P_SZ; k0 += 4) {
                const int kk = k0 + 2 * kgrp;
                const v2f av0 = *(const v2f*)(lrow0 + IN_DIM + kk);
                const v2f av1 = *(const v2f*)(lrow1 + IN_DIM + kk);
                const float* br = Wh + (size_t)kk * G4H;
                v2f b0; b0.x = br[nb];      b0.y = br[nb + G4H];
                v2f b1; b1.x = br[nb + 16]; b1.y = br[nb + 16 + G4H];
                a00 = wmma_f32(av0, b0, a00);
                a01 = wmma_f32(av0, b1, a01);
                a10 = wmma_f32(av1, b0, a10);
                a11 = wmma_f32(av1, b1, a11);
            }
#pragma unroll
            for (int r = 0; r < 8; ++r) {
                const int m = m_base + mrow0 + r + mhi;
                float* g0 = gates + (size_t)m * G4H + ncol0 + l15;
                float* g1 = gates + (size_t)(m + 16) * G4H + ncol0 + l15;
                g0[0]  = a00[r];
                g0[16] = a01[r];
                g1[0]  = a10[r];
                g1[16] = a11[r];
            }
        }
        grid_barrier(bar_count, bar_gen);

        // ---------- Phase 2: cell update (elementwise) ----------
        for (int idx = blockIdx.x * TPB + threadIdx.x; idx < B_SZ * H_SZ;
             idx += NUM_WG * TPB) {
            const int b  = idx / H_SZ;
            const int hu = idx - b * H_SZ;
            const float* grow = gates + (size_t)b * G4H;
            const float gi = grow[hu]            + bias[hu];
            const float gf = grow[H_SZ + hu]     + bias[H_SZ + hu];
            const float gg = grow[2 * H_SZ + hu] + bias[2 * H_SZ + hu];
            const float go = grow[3 * H_SZ + hu] + bias[3 * H_SZ + hu];
            const float iv = sigmoidf_(gi);
            const float fv = sigmoidf_(gf);
            const float gv = tanhf(gg);
            const float ov = sigmoidf_(go);
            const float cn = fv * c_state[idx] + iv * gv;
            c_state[idx] = cn;
            g2[idx] = ov * tanhf(cn);
        }
        grid_barrier(bar_count, bar_gen);

        // ---------- Phase 3: projection h = g2 @ Wp ----------
        if (pm >= 0) {
            v8f acc = {};
            const float* arow = g2 + (size_t)(pm + l15) * H_SZ;
#pragma unroll 2
            for (int k0 = 0; k0 < H_SZ; k0 += 4) {
                const int kk = k0 + 2 * kgrp;
                v2f a; a.x = arow[kk]; a.y = arow[kk + 1];
                const float* br = Wp + (size_t)kk * P_SZ + pn + l15;
                v2f b; b.x = br[0]; b.y = br[P_SZ];
                acc = wmma_f32(a, b, acc);
            }
#pragma unroll
            for (int r = 0; r < 8; ++r) {
                const int m = pm + r + mhi;
                const int n = pn + l15;
                const float v = acc[r];
                h_state[(size_t)m * P_SZ + n] = v;
                seq_out[(size_t)m * T_LEN * P_SZ + (size_t)t * P_SZ + n] = v;
            }
        }
        grid_barrier(bar_count, bar_gen);
    }
}

// Mean-pool over T, embeds = relu(pooled @ W^T + lb), then L2-normalize rows.
__global__ __launch_bounds__(P_SZ) void head_kernel(
    const float* __restrict__ seq,  // [B, T, P]
    const float* __restrict__ W,    // [P, P]
    const float* __restrict__ lb,   // [P]
    float* __restrict__ out)        // [B, P]
{
    __shared__ float pooled[P_SZ];
    __shared__ float red[P_SZ];
    const int b = blockIdx.x;
    const int j = threadIdx.x;

    float s = 0.0f;
    const float* base = seq + (size_t)b * T_LEN * P_SZ + j;
    for (int t = 0; t < T_LEN; ++t) s += base[(size_t)t * P_SZ];
    pooled[j] = s * (1.0f / (float)T_LEN);
    __syncthreads();

    float acc = lb[j];
    const float* wrow = W + (size_t)j * P_SZ;
    for (int k = 0; k < P_SZ; ++k) acc += pooled[k] * wrow[k];
    const float e = acc > 0.0f ? acc : 0.0f;

    red[j] = e * e;
    __syncthreads();
    for (int off = P_SZ / 2; off > 0; off >>= 1) {
        if (j < off) red[j] += red[j + off];
        __syncthreads();
    }
    const float nrm = sqrtf(red[0]);
    out[(size_t)b * P_SZ + j] = e / nrm;
}

extern "C" void kernel_launch(void* const* d_in, const int* in_sizes, int n_in,
                              void* d_out, int out_size, void* d_ws, size_t ws_size,
                              hipStream_t stream) {
    (void)in_sizes; (void)n_in; (void)out_size; (void)ws_size;

    const float* x   = (const float*)d_in[0];
    const float* Wx0 = (const float*)d_in[1];
    const float* Wh0 = (const float*)d_in[2];
    const float* b0  = (const float*)d_in[3];
    const float* Wp0 = (const float*)d_in[4];
    const float* Wx1 = (const float*)d_in[5];
    const float* Wh1 = (const float*)d_in[6];
    const float* b1  = (const float*)d_in[7];
    const float* Wp1 = (const float*)d_in[8];
    const float* Wx2 = (const float*)d_in[9];
    const float* Wh2 = (const float*)d_in[10];
    const float* b2  = (const float*)d_in[11];
    const float* Wp2 = (const float*)d_in[12];
    const float* W   = (const float*)d_in[13];
    const float* lb  = (const float*)d_in[14];
    float* out = (float*)d_out;

    // ---- workspace carve-up ----
    float* ws      = (float*)d_ws;
    float* c_state = ws;                                  // B*H
    float* h_state = c_state + (size_t)B_SZ * H_SZ;       // B*P
    float* gates   = h_state + (size_t)B_SZ * P_SZ;       // B*4H
    float* g2      = gates   + (size_t)B_SZ * G4H;        // B*H
    unsigned* bars = (unsigned*)(g2 + (size_t)B_SZ * H_SZ);
    float* seq0    = (float*)(bars + 16);                 // B*T*P
    float* seq1    = seq0 + (size_t)B_SZ * T_LEN * P_SZ;  // B*T*P

    const size_t sh80  = (size_t)64 * (I_DIM + P_SZ + 4) * sizeof(float); // 87 KB
    const size_t sh256 = (size_t)64 * (P_SZ + P_SZ + 4) * sizeof(float);  // 129 KB

    init_kernel<<<(B_SZ * H_SZ + 255) / 256, 256, 0, stream>>>(c_state, h_state, bars);

    lstmp_layer_kernel<I_DIM><<<NUM_WG, TPB, sh80, stream>>>(
        x, Wx0, Wh0, b0, Wp0, seq0, c_state, h_state, gates, g2, bars, bars + 1);
    lstmp_layer_kernel<P_SZ><<<NUM_WG, TPB, sh256, stream>>>(
        seq0, Wx1, Wh1, b1, Wp1, seq1, c_state, h_state, gates, g2, bars, bars + 1);
    lstmp_layer_kernel<P_SZ><<<NUM_WG, TPB, sh256, stream>>>(
        seq1, Wx2, Wh2, b2, Wp2, seq0, c_state, h_state, gates, g2, bars, bars + 1);

    head_kernel<<<B_SZ, P_SZ, 0, stream>>>(seq0, W, lb, out);
}